// ManifoldMatchingLoss_40638980554770
// MI455X (gfx1250) — compile-verified
//
#include <hip/hip_runtime.h>
#include <hip/hip_bf16.h>
#include <math.h>

typedef __attribute__((ext_vector_type(2))) float v2f;
typedef __attribute__((ext_vector_type(8))) float v8f;

#define BDIM 128  // batch size B

// ---------------------------------------------------------------------------
// Kernel 1: G = X * X^T, one 16x16 output tile per block, fp32 WMMA.
// Block = 128 threads = 4 wave32s; wave w accumulates K-slice w; partial
// accumulators reduced through LDS in a fixed order (deterministic).
// K-loop unrolled by 16 floats (4 WMMA chunks) with 2 independent
// accumulators: 8 loads issued before the first wait, 4 WMMAs per trip.
// D/4 must be divisible by 16 (12544/4=3136=196*16, 4096/4=1024=64*16).
// ---------------------------------------------------------------------------
__global__ __launch_bounds__(128) void gram_wmma_kernel(
    const float* __restrict__ X, int D, float* __restrict__ G) {
  const int lane = threadIdx.x & 31;
  const int wave = threadIdx.x >> 5;
  const int ti = blockIdx.x * 16;  // M tile base (rows of A)
  const int tj = blockIdx.y * 16;  // N tile base (rows of B, since B = X^T)

  // ISA VGPR layout for 32-bit 16x4 A / 4x16 B:
  //   lane holds row r = lane&15; VGPR pair covers K = (lane>>4)*2 .. +1
  //   within each 4-wide K chunk.
  const int r   = lane & 15;
  const int khi = (lane >> 4) * 2;

  const float* Ap = X + (size_t)(ti + r) * D + khi;
  const float* Bp = X + (size_t)(tj + r) * D + khi;

  // Exact per-wave K slice in float units (D/4 floats each, multiple of 16).
  const int slice = D >> 2;
  const int kbeg  = wave * slice;
  const int kend  = kbeg + slice;

  v8f c0 = {};
  v8f c1 = {};
  for (int k = kbeg; k < kend; k += 16) {
    // Prefetch one unroll step ahead (global_prefetch_b8).
    __builtin_prefetch(Ap + k + 16, 0, 1);
    __builtin_prefetch(Bp + k + 16, 0, 1);

    v2f a0 = *(const v2f*)(Ap + k);
    v2f b0 = *(const v2f*)(Bp + k);
    v2f a1 = *(const v2f*)(Ap + k + 4);
    v2f b1 = *(const v2f*)(Bp + k + 4);
    v2f a2 = *(const v2f*)(Ap + k + 8);
    v2f b2 = *(const v2f*)(Bp + k + 8);
    v2f a3 = *(const v2f*)(Ap + k + 12);
    v2f b3 = *(const v2f*)(Bp + k + 12);

    // 8 args: (neg_a, A, neg_b, B, c_mod, C, reuse_a, reuse_b)
    c0 = __builtin_amdgcn_wmma_f32_16x16x4_f32(
        false, a0, false, b0, (short)0, c0, false, false);
    c1 = __builtin_amdgcn_wmma_f32_16x16x4_f32(
        false, a1, false, b1, (short)0, c1, false, false);
    c0 = __builtin_amdgcn_wmma_f32_16x16x4_f32(
        false, a2, false, b2, (short)0, c0, false, false);
    c1 = __builtin_amdgcn_wmma_f32_16x16x4_f32(
        false, a3, false, b3, (short)0, c1, false, false);
  }

  // Reduce the 4 per-wave partial tiles through LDS (fixed order).
  __shared__ float part[4][8][32];
#pragma unroll
  for (int v = 0; v < 8; ++v) part[wave][v][lane] = c0[v] + c1[v];
  __syncthreads();

  if (wave == 0) {
    // C/D layout: VGPR v, lane L -> M = v + (L>>4)*8, N = L&15.
    const int nc = tj + (lane & 15);
    const int mb = ti + (lane >> 4) * 8;
#pragma unroll
    for (int v = 0; v < 8; ++v) {
      float s = (part[0][v][lane] + part[1][v][lane]) +
                (part[2][v][lane] + part[3][v][lane]);
      G[(size_t)(mb + v) * BDIM + nc] = s;
    }
  }
}

// ---------------------------------------------------------------------------
// Block-wide reductions over 128 threads (tree in LDS, fixed order).
// ---------------------------------------------------------------------------
__device__ inline float block_max128(float v, float* red) {
  const int t = threadIdx.x;
  red[t] = v;
  __syncthreads();
  for (int s = 64; s >= 1; s >>= 1) {
    if (t < s) red[t] = fmaxf(red[t], red[t + s]);
    __syncthreads();
  }
  float r = red[0];
  __syncthreads();
  return r;
}

__device__ inline float block_sum128(float v, float* red) {
  const int t = threadIdx.x;
  red[t] = v;
  __syncthreads();
  for (int s = 64; s >= 1; s >>= 1) {
    if (t < s) red[t] = red[t] + red[t + s];
    __syncthreads();
  }
  float r = red[0];
  __syncthreads();
  return r;
}

// ---------------------------------------------------------------------------
// Kernel 2: fused distance -> log_softmax -> KL row reduction.
// One block per row i, thread j handles column j.
//   dist[i,j] = sqrt(max(0, G[i,i] + G[j,j] - 2 G[i,j]))  (0 where sq<=0)
//   A = log_softmax(-dist/T, axis=1)
//   out[i] = sum_j exp(Ag) * (Ag - Af)
// ---------------------------------------------------------------------------
__global__ __launch_bounds__(BDIM) void kl_rows_kernel(
    const float* __restrict__ Gf, const float* __restrict__ Gg,
    const float* __restrict__ temperature, float* __restrict__ out) {
  __shared__ float red[BDIM];
  const int i = blockIdx.x;
  const int j = threadIdx.x;
  const float invT = 1.0f / temperature[0];

  float sq = Gf[i * BDIM + i] + Gf[j * BDIM + j] - 2.0f * Gf[i * BDIM + j];
  const float xf = -(sq > 0.0f ? sqrtf(sq) : 0.0f) * invT;
  sq = Gg[i * BDIM + i] + Gg[j * BDIM + j] - 2.0f * Gg[i * BDIM + j];
  const float xg = -(sq > 0.0f ? sqrtf(sq) : 0.0f) * invT;

  // Stable row log-softmax for both matrices.
  const float mf  = block_max128(xf, red);
  const float lf  = mf + logf(block_sum128(expf(xf - mf), red));
  const float mg  = block_max128(xg, red);
  const float lg  = mg + logf(block_sum128(expf(xg - mg), red));

  const float af = xf - lf;
  const float ag = xg - lg;
  const float kl = expf(ag) * (ag - af);

  const float tot = block_sum128(kl, red);
  if (j == 0) out[i] = tot;
}

// ---------------------------------------------------------------------------
// Launch: gram(image) -> Gf, gram(lang) -> Gg, fused KL epilogue.
// Workspace: 2 * 128*128 fp32 = 128 KB (L2-resident on MI455X).
// ---------------------------------------------------------------------------
extern "C" void kernel_launch(void* const* d_in, const int* in_sizes, int n_in,
                              void* d_out, int out_size, void* d_ws, size_t ws_size,
                              hipStream_t stream) {
  const float* image_feats = (const float*)d_in[0];  // (128, 49, 256) fp32
  const float* lang_feats  = (const float*)d_in[1];  // (128, 16, 256) fp32
  const float* temperature = (const float*)d_in[2];  // scalar fp32

  const int Dimg = in_sizes[0] / BDIM;  // 12544
  const int Dlng = in_sizes[1] / BDIM;  // 4096

  float* Gf = (float*)d_ws;
  float* Gg = Gf + BDIM * BDIM;

  dim3 tiles(BDIM / 16, BDIM / 16);  // 8x8 tiles of 16x16
  gram_wmma_kernel<<<tiles, 128, 0, stream>>>(image_feats, Dimg, Gf);
  gram_wmma_kernel<<<tiles, 128, 0, stream>>>(lang_feats, Dlng, Gg);

  kl_rows_kernel<<<BDIM, BDIM, 0, stream>>>(Gf, Gg, temperature, (float*)d_out);
}